// GCNEncoder_47047071760639
// MI455X (gfx1250) — compile-verified
//
#include <hip/hip_runtime.h>
#include <hip/hip_bf16.h>

typedef float v2f __attribute__((ext_vector_type(2)));
typedef float v8f __attribute__((ext_vector_type(8)));

#define GCN_F 128       // GEMM output width (HIDDEN, and OUT_MU|OUT_LV packed)
#define BS_STRIDE 136   // LDS row stride (words): 2*136 % 64 == 16 -> conflict-free halves

// ---------------- degree / normalization ----------------

__global__ void gcn_zero_i32(int* p, int n) {
    int i = blockIdx.x * blockDim.x + threadIdx.x;
    if (i < n) p[i] = 0;
}

__global__ void gcn_count_deg(const int* __restrict__ col, int* deg, int e) {
    int i = blockIdx.x * blockDim.x + threadIdx.x;
    if (i < e) atomicAdd(&deg[col[i]], 1);
}

__global__ void gcn_dinv(const int* __restrict__ deg, float* dinv, int n) {
    int i = blockIdx.x * blockDim.x + threadIdx.x;
    if (i < n) dinv[i] = rsqrtf((float)(deg[i] + 1));   // +1 self loop
}

// ---------------- exclusive scan (3-phase) ----------------

__global__ void gcn_scan_block(const int* __restrict__ deg, int* __restrict__ off,
                               int* __restrict__ bsum, int n) {
    __shared__ int s[256];
    int i = blockIdx.x * 256 + threadIdx.x;
    int v = (i < n) ? deg[i] : 0;
    s[threadIdx.x] = v;
    __syncthreads();
    for (int d = 1; d < 256; d <<= 1) {
        int t = (threadIdx.x >= d) ? s[threadIdx.x - d] : 0;
        __syncthreads();
        s[threadIdx.x] += t;
        __syncthreads();
    }
    if (i < n) off[i] = s[threadIdx.x] - v;            // exclusive
    if (threadIdx.x == 255) bsum[blockIdx.x] = s[255];
}

__global__ void gcn_scan_bsum(int* bsum, int nb) {
    if (threadIdx.x == 0 && blockIdx.x == 0) {
        int run = 0;
        for (int b = 0; b < nb; ++b) { int t = bsum[b]; bsum[b] = run; run += t; }
    }
}

__global__ void gcn_scan_add(int* off, const int* __restrict__ bsum, int n) {
    int i = blockIdx.x * 256 + threadIdx.x;
    if (i < n) off[i] += bsum[blockIdx.x];
}

__global__ void gcn_copy_i32(const int* __restrict__ src, int* __restrict__ dst, int n) {
    int i = blockIdx.x * blockDim.x + threadIdx.x;
    if (i < n) dst[i] = src[i];
}

// CSR fill: one int atomic per edge
__global__ void gcn_fill_csr(const int* __restrict__ rows, const int* __restrict__ cols,
                             int* cursor, int* __restrict__ srcList, int e) {
    int i = blockIdx.x * blockDim.x + threadIdx.x;
    if (i < e) {
        int c = cols[i];
        int p = atomicAdd(&cursor[c], 1);
        srcList[p] = rows[i];
    }
}

// Pack Wmu[128,64] | Wlv[128,64] -> W23[128,128] row-major
__global__ void gcn_pack_w23(const float* __restrict__ Wmu,
                             const float* __restrict__ Wlv,
                             float* __restrict__ W23) {
    int i = blockIdx.x * blockDim.x + threadIdx.x;   // 128*128 threads
    int k = i >> 7, n = i & 127;
    W23[i] = (n < 64) ? Wmu[k * 64 + n] : Wlv[k * 64 + (n - 64)];
}

// ---------------- WMMA f32 GEMM, LDS-staged B, row-scale epilogue ----------------
// C[m, 0:128] = rowScale[m] * (A[m, 0:K] @ B[0:K, 0:128]);  one wave -> 16x128 tile
__global__ __launch_bounds__(256)
void gcn_gemm_rowscale_wmma(const float* __restrict__ A,
                            const float* __restrict__ B,
                            const float* __restrict__ rowScale,
                            float* __restrict__ C,
                            int M, int K) {
    __shared__ float Bs[64 * BS_STRIDE];               // 34.8 KB

    const int tid  = threadIdx.x;
    const int lane = tid & 31;
    const int wave = tid >> 5;
    const int rt   = blockIdx.x * 8 + wave;            // 16-row tile index
    const bool active = (rt * 16 < M);
    const int m0   = rt * 16;
    const int half = lane >> 4;                        // 0: K+{0,1}, 1: K+{2,3}
    const int l16  = lane & 15;

    const float* Arow = active ? (A + (size_t)(m0 + l16) * K) : A;

    v8f acc[8] = {};

    for (int kb = 0; kb < K; kb += 64) {
        __syncthreads();
        // stage B[kb..kb+64) x 128 -> LDS (padded rows), 256 threads x 8 float4
        for (int idx = tid; idx < 64 * 32; idx += 256) {
            int r = idx >> 5, c4 = idx & 31;
            float4 v = ((const float4*)(B + (size_t)(kb + r) * GCN_F))[c4];
            *(float4*)(Bs + r * BS_STRIDE + c4 * 4) = v;
        }
        __syncthreads();

        if (active) {
            for (int kl = 0; kl < 64; kl += 4) {
                const int ka = kl + 2 * half;
                v2f a;
                a.x = Arow[kb + ka];
                a.y = Arow[kb + ka + 1];
                const float* B0 = Bs + ka * BS_STRIDE + l16;
#pragma unroll
                for (int j = 0; j < 8; ++j) {
                    v2f b;
                    b.x = B0[16 * j];
                    b.y = B0[16 * j + BS_STRIDE];
                    acc[j] = __builtin_amdgcn_wmma_f32_16x16x4_f32(
                        false, a, false, b, (short)0, acc[j], false, false);
                }
            }
        }
    }

    if (active) {
#pragma unroll
        for (int v = 0; v < 8; ++v) {
            const int row = m0 + v + 8 * half;         // D: VGPR v -> row v + 8*half
            const float s = rowScale[row];
            float* Crow = C + (size_t)row * GCN_F + l16;
#pragma unroll
            for (int j = 0; j < 8; ++j) Crow[16 * j] = acc[j][v] * s;
        }
    }
}

// ---------------- CSR gather kernels (one wave per node, lane = float4 chunk) ----

// H[c] = relu(dinv[c] * (Ts[c] + sum_{e in in(c)} Ts[src]) + b1)
__global__ __launch_bounds__(256)
void gcn_gather_relu(const float* __restrict__ Ts, const int* __restrict__ srcList,
                     const int* __restrict__ off, const int* __restrict__ deg,
                     const float* __restrict__ dinv, const float* __restrict__ b1,
                     float* __restrict__ H, int nNodes) {
    const int node = blockIdx.x * 8 + (threadIdx.x >> 5);
    if (node >= nNodes) return;
    const int lane = threadIdx.x & 31;

    float4 acc = ((const float4*)(Ts + (size_t)node * GCN_F))[lane];  // self loop
    const int s = off[node];
    const int d = deg[node];
    for (int e = 0; e < d; ++e) {
        const int src = srcList[s + e];
        const float4 v = ((const float4*)(Ts + (size_t)src * GCN_F))[lane];
        acc.x += v.x; acc.y += v.y; acc.z += v.z; acc.w += v.w;
    }
    const float dv = dinv[node];
    const float4 bb = ((const float4*)b1)[lane];
    float4 r;
    r.x = fmaxf(dv * acc.x + bb.x, 0.0f);
    r.y = fmaxf(dv * acc.y + bb.y, 0.0f);
    r.z = fmaxf(dv * acc.z + bb.z, 0.0f);
    r.w = fmaxf(dv * acc.w + bb.w, 0.0f);
    ((float4*)(H + (size_t)node * GCN_F))[lane] = r;
}

// mu/lv = dinv[c] * (Ts[c] + sum Ts[src]) + bias, split-written to d_out
__global__ __launch_bounds__(256)
void gcn_gather_out(const float* __restrict__ Ts, const int* __restrict__ srcList,
                    const int* __restrict__ off, const int* __restrict__ deg,
                    const float* __restrict__ dinv, const float* __restrict__ bmu,
                    const float* __restrict__ blv, float* __restrict__ out, int nNodes) {
    const int node = blockIdx.x * 8 + (threadIdx.x >> 5);
    if (node >= nNodes) return;
    const int lane = threadIdx.x & 31;

    float4 acc = ((const float4*)(Ts + (size_t)node * GCN_F))[lane];  // self loop
    const int s = off[node];
    const int d = deg[node];
    for (int e = 0; e < d; ++e) {
        const int src = srcList[s + e];
        const float4 v = ((const float4*)(Ts + (size_t)src * GCN_F))[lane];
        acc.x += v.x; acc.y += v.y; acc.z += v.z; acc.w += v.w;
    }
    const float dv = dinv[node];
    float4 bb;
    float* dst;
    if (lane < 16) {                                   // mu: features 0..63
        bb  = ((const float4*)bmu)[lane];
        dst = out + (size_t)node * 64 + lane * 4;
    } else {                                           // lv: features 64..127
        bb  = ((const float4*)blv)[lane - 16];
        dst = out + (size_t)nNodes * 64 + (size_t)node * 64 + (lane - 16) * 4;
    }
    float4 r;
    r.x = dv * acc.x + bb.x;
    r.y = dv * acc.y + bb.y;
    r.z = dv * acc.z + bb.z;
    r.w = dv * acc.w + bb.w;
    *(float4*)dst = r;
}

// ---------------- launcher ----------------

extern "C" void kernel_launch(void* const* d_in, const int* in_sizes, int n_in,
                              void* d_out, int out_size, void* d_ws, size_t ws_size,
                              hipStream_t stream) {
    const float* X   = (const float*)d_in[0];
    const int*   ei  = (const int*)d_in[1];
    const float* W1  = (const float*)d_in[2];
    const float* b1  = (const float*)d_in[3];
    const float* Wmu = (const float*)d_in[4];
    const float* bmu = (const float*)d_in[5];
    const float* Wlv = (const float*)d_in[6];
    const float* blv = (const float*)d_in[7];
    float* out = (float*)d_out;

    const int N = in_sizes[0] / 256;   // 100000
    const int E = in_sizes[1] / 2;     // 1600000
    const int* rows = ei;              // edge_index[0] (sources)
    const int* cols = ei + E;          // edge_index[1] (targets)

    // workspace layout (16B-aligned float buffers first)
    float* Ts   = (float*)d_ws;                      // [N,128]
    float* Hbuf = Ts + (size_t)N * GCN_F;            // [N,128]
    float* W23  = Hbuf + (size_t)N * GCN_F;          // 128*128
    float* dinv = W23 + GCN_F * GCN_F;               // N
    int* degInt  = (int*)(dinv + N);                 // N
    int* off     = degInt + N;                       // N
    int* cursor  = off + N;                          // N
    int* bsum    = cursor + N;                       // nb
    const int nb = (N + 255) / 256;                  // 391
    int* srcList = bsum + ((nb + 3) & ~3);           // E

    // ---- CSR build + normalization
    gcn_zero_i32<<<(N + 255) / 256, 256, 0, stream>>>(degInt, N);
    gcn_count_deg<<<(E + 255) / 256, 256, 0, stream>>>(cols, degInt, E);
    gcn_dinv<<<(N + 255) / 256, 256, 0, stream>>>(degInt, dinv, N);
    gcn_scan_block<<<nb, 256, 0, stream>>>(degInt, off, bsum, N);
    gcn_scan_bsum<<<1, 64, 0, stream>>>(bsum, nb);
    gcn_scan_add<<<nb, 256, 0, stream>>>(off, bsum, N);
    gcn_copy_i32<<<(N + 255) / 256, 256, 0, stream>>>(off, cursor, N);
    gcn_fill_csr<<<(E + 255) / 256, 256, 0, stream>>>(rows, cols, cursor, srcList, E);
    gcn_pack_w23<<<64, 256, 0, stream>>>(Wmu, Wlv, W23);

    const int rowTiles = (N + 15) / 16;              // 6250
    const int gemmBlocks = (rowTiles + 7) / 8;       // 8 row tiles per block
    const int gatherBlocks = (N + 7) / 8;            // 8 nodes per block

    // ---- conv1: Ts = dinv*(X@W1); H = relu(dinv*(Ts[c] + gather) + b1)
    gcn_gemm_rowscale_wmma<<<gemmBlocks, 256, 0, stream>>>(X, W1, dinv, Ts, N, 256);
    gcn_gather_relu<<<gatherBlocks, 256, 0, stream>>>(Ts, srcList, off, degInt,
                                                      dinv, b1, Hbuf, N);

    // ---- conv2+3 fused: Ts = dinv*(H@[Wmu|Wlv]); mu/lv = dinv*(Ts[c]+gather)+bias
    gcn_gemm_rowscale_wmma<<<gemmBlocks, 256, 0, stream>>>(Hbuf, W23, dinv, Ts, N, 128);
    gcn_gather_out<<<gatherBlocks, 256, 0, stream>>>(Ts, srcList, off, degInt,
                                                     dinv, bmu, blv, out, N);
}